// FullNet_50826642981495
// MI455X (gfx1250) — compile-verified
//
#include <hip/hip_runtime.h>
#include <hip/hip_bf16.h>

typedef __attribute__((ext_vector_type(2))) float v2f;
typedef __attribute__((ext_vector_type(8))) float v8f;

__device__ __forceinline__ float softplusf(float v) {
    // numerically stable log(1+exp(v))
    return fmaxf(v, 0.0f) + log1pf(expf(-fabsf(v)));
}

// ---------------------------------------------------------------------------
// Generic direct convolution via V_WMMA_F32_16X16X4_F32 (exact fp32).
// Each wave computes a 32-pixel x 16-channel output tile as two 16x16 WMMA
// accumulators sharing one B (weight) fragment per K-chunk.
//   D[m][n] = sum_k A[m][k] * B[k][n],  m = consecutive output x-pixels,
//   n = output channels, k = flattened (ci,ky,kx) tap, K padded to mult 4.
// A layout (ISA 7.12.2, 32-bit A 16x4): lanes 0-15 hold taps k0,k0+1 for
//   row M=lane; lanes 16-31 hold taps k0+2,k0+3 for M=lane-16.
// B layout (mirror of documented sparse-B pattern): lane l holds column
//   N=l&15, rows (k0,k0+1) for l<16 and (k0+2,k0+3) for l>=16.
// C/D layout: VGPR r, lanes 0-15 -> M=r, lanes 16-31 -> M=r+8, N=lane&15.
// Branch-free K-loop: weights are zero-padded in LDS for k>=K and n>=Co, and
// offs[k]=0 for pad taps points at a valid finite input -> pad terms are
// exactly finite*0 = 0. act: 0 = none, 1 = softplus, 2 = leaky_relu(0.01)
// ---------------------------------------------------------------------------
__global__ void __launch_bounds__(128)
conv_wmma_kernel(const float* __restrict__ in, int Cin, int Hin, int Win,
                 const float* __restrict__ wgt, const float* __restrict__ bias,
                 int KH, int KW, int dil, int y0, int x0,
                 float* __restrict__ out, int Co, int ch0,
                 int Hout, int Wout, int act)
{
    extern __shared__ float smem[];
    const int K    = Cin * KH * KW;
    const int Kpad = (K + 3) & ~3;
    const int WROW = Kpad + 1;                 // odd stride -> LDS conflict-free
    float* lds_w = smem;                       // [16][WROW] zero-padded weights
    float* lds_b = smem + 16 * WROW;           // [16] bias
    int*   offs  = (int*)(lds_b + 16);         // [Kpad] per-tap input offsets

    const int nbase = blockIdx.y << 4;
    const int KWH   = KH * KW;

    for (int idx = threadIdx.x; idx < 16 * Kpad; idx += blockDim.x) {
        int n = idx / Kpad, k = idx - n * Kpad;
        int co = nbase + n;
        lds_w[n * WROW + k] = (co < Co && k < K) ? wgt[co * K + k] : 0.0f;
    }
    for (int k = threadIdx.x; k < Kpad; k += blockDim.x) {
        int o = 0;
        if (k < K) {
            int ci = k / KWH, r = k - ci * KWH;
            int ky = r / KW,  kx = r - ky * KW;
            o = (ci * Hin + ky * dil) * Win + kx * dil;
        }
        offs[k] = o;
    }
    if (threadIdx.x < 16) {
        int co = nbase + threadIdx.x;
        lds_b[threadIdx.x] = (co < Co) ? bias[co] : 0.0f;
    }
    __syncthreads();

    const int lane   = threadIdx.x & 31;
    const int half   = lane >> 4;
    const int lm     = lane & 15;
    const int tilesX = (Wout + 31) >> 5;       // 32 pixels per wave-tile
    const int tile   = (blockIdx.x << 2) + (threadIdx.x >> 5);
    if (tile >= Hout * tilesX) return;         // wave-uniform exit (EXEC all-1
    const int ty = tile / tilesX;              // for every WMMA)
    const int xb = (tile - ty * tilesX) << 5;

    int px0 = xb + lm;                         // clamp read lanes; stores are
    int px1 = xb + 16 + lm;                    // guarded at the end
    if (px0 >= Wout) px0 = Wout - 1;
    if (px1 >= Wout) px1 = Wout - 1;
    const float* pin0 = in + (ty + y0) * Win + (px0 + x0);
    const float* pin1 = in + (ty + y0) * Win + (px1 + x0);
    const float* wrow = lds_w + lm * WROW;

    v8f acc0 = {};
    v8f acc1 = {};
    #pragma unroll 2
    for (int k0 = 0; k0 < Kpad; k0 += 4) {
        const int t0 = k0 + (half << 1);
        const int o0 = offs[t0];
        const int o1 = offs[t0 + 1];
        v2f a0, a1, b;
        a0.x = pin0[o0];  a0.y = pin0[o1];
        a1.x = pin1[o0];  a1.y = pin1[o1];
        b.x  = wrow[t0];  b.y  = wrow[t0 + 1];
        acc0 = __builtin_amdgcn_wmma_f32_16x16x4_f32(false, a0, false, b,
                                                     (short)0, acc0, false, false);
        acc1 = __builtin_amdgcn_wmma_f32_16x16x4_f32(false, a1, false, b,
                                                     (short)0, acc1, false, false);
    }

    const int n = nbase + lm;
    if (n >= Co) return;
    const float bv = lds_b[lm];
    float* po = out + ((long)(ch0 + n) * Hout + ty) * (long)Wout;
    #pragma unroll
    for (int r = 0; r < 8; ++r) {
        const int xo = xb + (half << 3) + r;
        if (xo < Wout) {
            float v = acc0[r] + bv;
            if      (act == 1) v = softplusf(v);
            else if (act == 2) v = (v >= 0.0f) ? v : 0.01f * v;
            po[xo] = v;
        }
        const int xo1 = xo + 16;
        if (xo1 < Wout) {
            float v = acc1[r] + bv;
            if      (act == 1) v = softplusf(v);
            else if (act == 2) v = (v >= 0.0f) ? v : 0.01f * v;
            po[xo1] = v;
        }
    }
}

// dst[c][y][x] = src[c][y+2][x+2]  (channel-skip crop into a concat buffer)
__global__ void crop_copy_kernel(const float* __restrict__ src, int C, int Hs, int Ws,
                                 float* __restrict__ dst, int Hd, int Wd)
{
    long i   = (long)blockIdx.x * blockDim.x + threadIdx.x;
    long tot = (long)C * Hd * Wd;
    if (i >= tot) return;
    int  x = (int)(i % Wd);
    long t = i / Wd;
    int  y = (int)(t % Hd);
    int  c = (int)(t / Hd);
    dst[i] = src[((long)c * Hs + (y + 2)) * Ws + (x + 2)];
}

// P[s][c][yy][xx] = mean_{u,v<16} G[c][32*yy + 2u + (s>>1)][32*xx + 2v + (s&1)]
// G: (30,1024,1024), P: (4,30,32,32)
__global__ void pool_phase_kernel(const float* __restrict__ g, float* __restrict__ P)
{
    int i = blockIdx.x * blockDim.x + threadIdx.x;
    if (i >= 4 * 30 * 32 * 32) return;
    int xx = i & 31;
    int t  = i >> 5;
    int yy = t & 31;  t >>= 5;
    int c  = t % 30;
    int s  = t / 30;
    int py = s >> 1, pxp = s & 1;
    const float* base = g + ((long)c * 1024 + 32 * yy + py) * 1024 + 32 * xx + pxp;
    float sum = 0.0f;
    for (int u = 0; u < 16; ++u) {
        const float* row = base + (long)u * 2 * 1024;
        #pragma unroll
        for (int v = 0; v < 16; ++v) sum += row[2 * v];
    }
    P[i] = sum * (1.0f / 256.0f);
}

// GP[p][c*4+j][yy][xx] = P[perm[p][j]][c][yy][xx];  GP: (4,120,32,32)
__global__ void permute_kernel(const float* __restrict__ P, float* __restrict__ GP)
{
    int i = blockIdx.x * blockDim.x + threadIdx.x;
    if (i >= 4 * 120 * 32 * 32) return;
    int pix = i & 1023;
    int t   = i >> 10;
    int c4  = t % 120;
    int p   = t / 120;
    int c = c4 >> 2, j = c4 & 3;
    const int perm[4][4] = {{0,1,2,3},{1,0,3,2},{2,3,0,1},{3,2,1,0}};
    int s = perm[p][j];
    GP[i] = P[((s * 30 + c) << 10) + pix];
}

// grouped 1x1 conv on (4,Ci,32,32) -> (4,Co,32,32); act: 1 = softplus
__global__ void gconv_kernel(const float* __restrict__ in, int Ci,
                             const float* __restrict__ wgt, const float* __restrict__ bias,
                             float* __restrict__ out, int Co, int groups, int act)
{
    int i   = blockIdx.x * blockDim.x + threadIdx.x;
    int tot = 4 * Co * 1024;
    if (i >= tot) return;
    int pix = i & 1023;
    int t   = i >> 10;
    int o   = t % Co;
    int bb  = t / Co;
    int cig  = Ci / groups;
    int cog  = Co / groups;
    int gidx = o / cog;
    const float* ib = in + ((long)bb * Ci + gidx * cig) * 1024 + pix;
    const float* wr = wgt + o * cig;
    float s = bias[o];
    for (int k = 0; k < cig; ++k) s = fmaf(wr[k], ib[(long)k * 1024], s);
    out[i] = act ? softplusf(s) : s;
}

// final 45->4 conv + log_softmax over the 4 channels; out: (4,4,32,32)
__global__ void final_lsm_kernel(const float* __restrict__ in,
                                 const float* __restrict__ wgt,
                                 const float* __restrict__ bias,
                                 float* __restrict__ out)
{
    int i = blockIdx.x * blockDim.x + threadIdx.x;
    if (i >= 4 * 1024) return;
    int pix = i & 1023, bb = i >> 10;
    const float* ib = in + (long)bb * 45 * 1024 + pix;
    float v[4];
    #pragma unroll
    for (int o = 0; o < 4; ++o) {
        float s = bias[o];
        const float* wr = wgt + o * 45;
        for (int k = 0; k < 45; ++k) s = fmaf(wr[k], ib[k * 1024], s);
        v[o] = s;
    }
    float m = fmaxf(fmaxf(v[0], v[1]), fmaxf(v[2], v[3]));
    float sum = 0.0f;
    #pragma unroll
    for (int o = 0; o < 4; ++o) sum += expf(v[o] - m);
    float lse = m + logf(sum);
    #pragma unroll
    for (int o = 0; o < 4; ++o) out[((long)(bb * 4 + o) << 10) + pix] = v[o] - lse;
}

// ---------------------------------------------------------------------------
extern "C" void kernel_launch(void* const* d_in, const int* in_sizes, int n_in,
                              void* d_out, int out_size, void* d_ws, size_t ws_size,
                              hipStream_t stream)
{
    (void)in_sizes; (void)n_in; (void)out_size; (void)ws_size;
    const float* x = (const float*)d_in[0];          // (3,1032,1032)
    auto Wp = [&](int i) { return (const float*)d_in[i]; };
    float* ws = (float*)d_ws;

    // workspace regions (floats); lifetimes are stream-ordered & reuse-safe.
    float* Y   = ws;                 // 26.6M: y1 (25,1030,1030) then y2 (25,1026,1026)
    float* X1  = ws + 27000000;      // 56.1M: (53,1028,1028)
    float* Hh  = ws + 84000000;      // 108.0M: (103,1024,1024)
    float* ABC = ws;                 // 62.9M: a|b|c (60,1024,1024) — Y,X1 dead by then
    float* Gg  = ws + 63000000;      // 31.5M: (30,1024,1024)    — overlaps dead Hh head
    float* Pp  = ws + 95000000;      // (4,30,32,32)
    float* GP  = ws + 95200000;      // (4,120,32,32)
    float* T0  = ws + 95800000;      // (4,180,32,32)
    float* T1  = ws + 96600000;      // (4,90,32,32)
    float* T2  = ws + 97000000;      // (4,90,32,32)
    float* T3  = ws + 97400000;      // (4,45,32,32)
    float* T4  = ws + 97600000;      // (4,45,32,32)

    auto convL = [&](const float* in, int Cin, int Hin, int Win, int wi,
                     int KH, int KW, int dil, int y0, int x0,
                     float* outp, int Co, int ch0, int Hout, int Wout, int act) {
        int K    = Cin * KH * KW;
        int Kpad = (K + 3) & ~3;
        int tilesX = (Wout + 31) / 32;
        int tiles  = Hout * tilesX;
        dim3 grid((tiles + 3) / 4, (Co + 15) / 16);
        size_t shmem = (size_t)(16 * (Kpad + 1) + 16 + Kpad) * sizeof(float);
        conv_wmma_kernel<<<grid, 128, shmem, stream>>>(
            in, Cin, Hin, Win, Wp(wi), Wp(wi + 1),
            KH, KW, dil, y0, x0, outp, Co, ch0, Hout, Wout, act);
    };
    auto crop = [&](const float* src, int C, int Hs, int Ws, float* dst, int Hd, int Wd) {
        long tot = (long)C * Hd * Wd;
        crop_copy_kernel<<<dim3((unsigned)((tot + 255) / 256)), 256, 0, stream>>>(
            src, C, Hs, Ws, dst, Hd, Wd);
    };
    auto gconv = [&](const float* in, int Ci, int wi, float* outp, int Co, int groups, int act) {
        int tot = 4 * Co * 1024;
        gconv_kernel<<<(tot + 255) / 256, 256, 0, stream>>>(
            in, Ci, Wp(wi), Wp(wi + 1), outp, Co, groups, act);
    };

    // ---- stage 1 dfd (input x: 3x1032x1032) ----
    // y1 = softplus([h1|v1|f1])  -> (25,1030,1030)
    convL(x, 3, 1032, 1032,  1, 1, 3, 1, 1, 0, Y, 10,  0, 1030, 1030, 1);
    convL(x, 3, 1032, 1032,  3, 3, 1, 1, 0, 1, Y, 10, 10, 1030, 1030, 1);
    convL(x, 3, 1032, 1032,  5, 3, 3, 1, 0, 0, Y,  5, 20, 1030, 1030, 1);
    // x1 = [crop(x) | hd|vd|fd | h2|v2|f2] -> (53,1028,1028)
    crop(x, 3, 1032, 1032, X1, 1028, 1028);
    convL(x, 3, 1032, 1032, 13, 1, 3, 2, 2, 0, X1, 10,  3, 1028, 1028, 0);
    convL(x, 3, 1032, 1032, 15, 3, 1, 2, 0, 2, X1, 10, 13, 1028, 1028, 0);
    convL(x, 3, 1032, 1032, 17, 3, 3, 2, 0, 0, X1,  5, 23, 1028, 1028, 0);
    convL(Y, 25, 1030, 1030, 7, 1, 3, 1, 1, 0, X1, 10, 28, 1028, 1028, 0);
    convL(Y, 25, 1030, 1030, 9, 3, 1, 1, 0, 1, X1, 10, 38, 1028, 1028, 0);
    convL(Y, 25, 1030, 1030,11, 3, 3, 1, 0, 0, X1,  5, 48, 1028, 1028, 0);

    // ---- stage 2 dfd (input x1: 53x1028x1028) ----
    // y2 -> (25,1026,1026), reuses Y region
    convL(X1, 53, 1028, 1028, 19, 1, 3, 1, 1, 0, Y, 10,  0, 1026, 1026, 1);
    convL(X1, 53, 1028, 1028, 21, 3, 1, 1, 0, 1, Y, 10, 10, 1026, 1026, 1);
    convL(X1, 53, 1028, 1028, 23, 3, 3, 1, 0, 0, Y,  5, 20, 1026, 1026, 1);
    // h = [crop(x1) | hd|vd|fd | h2|v2|f2] -> (103,1024,1024)
    crop(X1, 53, 1028, 1028, Hh, 1024, 1024);
    convL(X1, 53, 1028, 1028, 31, 1, 3, 2, 2, 0, Hh, 10, 53, 1024, 1024, 0);
    convL(X1, 53, 1028, 1028, 33, 3, 1, 2, 0, 2, Hh, 10, 63, 1024, 1024, 0);
    convL(X1, 53, 1028, 1028, 35, 3, 3, 2, 0, 0, Hh,  5, 73, 1024, 1024, 0);
    convL(Y,  25, 1026, 1026, 25, 1, 3, 1, 1, 0, Hh, 10, 78, 1024, 1024, 0);
    convL(Y,  25, 1026, 1026, 27, 3, 1, 1, 0, 1, Hh, 10, 88, 1024, 1024, 0);
    convL(Y,  25, 1026, 1026, 29, 3, 3, 1, 0, 0, Hh,  5, 98, 1024, 1024, 0);

    // ---- pixelwise 1x1 chain (a|b|c packed contiguously in ABC, 60ch) ----
    convL(Hh, 103, 1024, 1024, 37, 1, 1, 1, 0, 0, ABC, 30,  0, 1024, 1024, 2);
    convL(ABC, 30, 1024, 1024, 39, 1, 1, 1, 0, 0, ABC, 15, 30, 1024, 1024, 2);
    convL(ABC, 45, 1024, 1024, 41, 1, 1, 1, 0, 0, ABC, 15, 45, 1024, 1024, 2);
    convL(ABC, 60, 1024, 1024, 43, 1, 1, 1, 0, 0, Gg,  30,  0, 1024, 1024, 0);

    // ---- phase pool + permutation assembly ----
    pool_phase_kernel<<<(4 * 30 * 32 * 32 + 255) / 256, 256, 0, stream>>>(Gg, Pp);
    permute_kernel<<<(4 * 120 * 32 * 32 + 255) / 256, 256, 0, stream>>>(Pp, GP);

    // ---- blockwise grouped 1x1 convs ----
    gconv(GP, 120, 45, T0, 180, 30, 1);
    gconv(T0, 180, 47, T1,  90, 30, 1);
    gconv(T1,  90, 49, T2,  90, 30, 1);
    gconv(T2,  90, 51, T3,  45,  1, 1);
    gconv(T3,  45, 53, T4,  45,  1, 1);
    final_lsm_kernel<<<(4 * 1024 + 255) / 256, 256, 0, stream>>>(
        T4, Wp(55), Wp(56), (float*)d_out);
}